// InteractionArch_63273458205226
// MI455X (gfx1250) — compile-verified
//
#include <hip/hip_runtime.h>

// DLRM interaction arch for MI455X (gfx1250, wave32, WMMA).
// out[b] = concat(dense[b] (128), triu_{k=1}(C_b C_b^T) (351)),  C_b = [dense_b; sparse_b] in R^{27x128}

typedef __attribute__((ext_vector_type(2))) float v2f;
typedef __attribute__((ext_vector_type(8))) float v8f;

#define DIM        128
#define NROWS      27
#define PADROWS    32
#define LDS_STRIDE 132      // floats per row; stride%64 == 4 -> conflict-free ds_load_b64 pattern
#define OUT_COLS   479      // 128 dense + 351 triu

__global__ __launch_bounds__(128) void dlrm_interact_wmma(
    const float* __restrict__ dense,
    const float* __restrict__ sparse,
    float* __restrict__ out)
{
    __shared__ float lds[PADROWS * LDS_STRIDE];   // 16896 B

    const int b   = blockIdx.x;
    const int tid = threadIdx.x;

    // ---- Stage C (27x128) into LDS, rows 27..31 zeroed. float4 coalesced global loads. ----
    const float4* drow = reinterpret_cast<const float4*>(dense  + (size_t)b * DIM);
    const float4* srow = reinterpret_cast<const float4*>(sparse + (size_t)b * (NROWS - 1) * DIM);
    #pragma unroll
    for (int i = tid; i < PADROWS * 32; i += 128) {     // 32 float4 per row
        const int row = i >> 5;
        const int c4  = i & 31;
        float4 v = make_float4(0.f, 0.f, 0.f, 0.f);
        if (row == 0)          v = drow[c4];
        else if (row < NROWS)  v = srow[(row - 1) * 32 + c4];
        *reinterpret_cast<float4*>(&lds[row * LDS_STRIDE + c4 * 4]) = v;
    }
    __syncthreads();

    const int wave = __builtin_amdgcn_readfirstlane(tid >> 5);  // scalar branch -> EXEC all-ones in WMMA region
    const int lane = tid & 31;

    if (wave < 3) {
        // Tile assignment over the symmetric 32x32 padded Gram: (I,J) in {(0,0),(0,1),(1,1)}
        const int I = (wave == 2) ? 1 : 0;
        const int J = (wave == 0) ? 0 : 1;

        // V_WMMA_F32_16X16X4_F32 fragment layout (ISA 7.12.2):
        //   A 16x4 : lane L -> row M = 16I + (L&15), VGPR{0,1} = K = k + 2*(L>>4) + {0,1}
        //   B 4x16 : lane L -> col N = 16J + (L&15), VGPR{0,1} = same K split
        // Both are contiguous float2 reads of a C row in LDS.
        const int rA   = I * 16 + (lane & 15);
        const int rB   = J * 16 + (lane & 15);
        const int koff = (lane >> 4) * 2;
        const float* pa = &lds[rA * LDS_STRIDE + koff];
        const float* pb = &lds[rB * LDS_STRIDE + koff];

        v8f acc = {};
        #pragma unroll
        for (int k = 0; k < DIM; k += 4) {
            v2f a  = *reinterpret_cast<const v2f*>(pa + k);   // ds_load_b64, bank-conflict-free
            v2f bb = *reinterpret_cast<const v2f*>(pb + k);
            // 8 args: (neg_a, A, neg_b, B, c_mod, C, reuse_a, reuse_b)
            acc = __builtin_amdgcn_wmma_f32_16x16x4_f32(
                false, a, false, bb, (short)0, acc, false, false);
        }

        // ---- Scatter strict upper triangle. C/D layout: VGPR r -> M = r + 8*(L>>4), N = L&15. ----
        float* orow = out + (size_t)b * OUT_COLS + DIM;
        const int n     = J * 16 + (lane & 15);
        const int mbase = I * 16 + ((lane >> 4) << 3);
        #pragma unroll
        for (int r = 0; r < 8; ++r) {
            const int m = mbase + r;
            if (m < n && n < NROWS) {
                const int idx = m * (NROWS - 1) - (m * (m - 1)) / 2 + (n - m - 1);
                orow[idx] = acc[r];
            }
        }
    } else {
        // Wave 3: pass dense features through to out[b][0:128]
        float* orow = out + (size_t)b * OUT_COLS;
        const float* drowf = dense + (size_t)b * DIM;
        #pragma unroll
        for (int i = 0; i < 4; ++i)
            orow[lane + 32 * i] = drowf[lane + 32 * i];
    }
}

extern "C" void kernel_launch(void* const* d_in, const int* in_sizes, int n_in,
                              void* d_out, int out_size, void* d_ws, size_t ws_size,
                              hipStream_t stream) {
    const float* dense  = (const float*)d_in[0];   // (B, 128) fp32
    const float* sparse = (const float*)d_in[1];   // (B, 26, 128) fp32
    float* out = (float*)d_out;                    // (B, 479) fp32
    const int B = in_sizes[0] / DIM;
    dlrm_interact_wmma<<<B, 128, 0, stream>>>(dense, sparse, out);
}